// GCN_88648124990117
// MI455X (gfx1250) — compile-verified
//
#include <hip/hip_runtime.h>
#include <hip/hip_bf16.h>

// ---------------------------------------------------------------------------
// GCN (2x GCNConv + mean-pool + linear) for MI455X (gfx1250).
// Memory/atomic-bound workload (~2 GB traffic vs 5 GFLOP dense) -> keep fp32,
// use V_WMMA_F32_16X16X4_F32 for the node-feature transforms.
// ---------------------------------------------------------------------------

#define N_NODES  100000
#define N_EDGES  600000
#define N_GRAPHS 2000
#define NCLS     4

typedef float v2f __attribute__((ext_vector_type(2)));
typedef float v8f __attribute__((ext_vector_type(8)));

// ---------------- utility kernels -----------------------------------------

__global__ void k_zero(float* __restrict__ p, int n) {
  int i = blockIdx.x * blockDim.x + threadIdx.x;
  if (i < n) p[i] = 0.0f;
}

// deg[dst] += 1 per edge (self-loop added in k_dinv)
__global__ void k_deg(const int* __restrict__ dst, float* __restrict__ deg) {
  int e = blockIdx.x * blockDim.x + threadIdx.x;
  if (e < N_EDGES) atomicAdd(&deg[dst[e]], 1.0f);
}

// in-place: deg -> deg^{-1/2}, with the +1 self loop (deg >= 1 always)
__global__ void k_dinv(float* __restrict__ deg) {
  int i = blockIdx.x * blockDim.x + threadIdx.x;
  if (i < N_NODES) deg[i] = rsqrtf(deg[i] + 1.0f);
}

// x0[n, 0:64] = emb_table[tok[n], 0:64]
__global__ void k_embed(const int* __restrict__ tok, const float* __restrict__ emb,
                        float* __restrict__ x0) {
  size_t idx = (size_t)blockIdx.x * blockDim.x + threadIdx.x;   // < N*64
  int node = (int)(idx >> 6);
  int c    = (int)(idx & 63);
  x0[idx] = emb[((size_t)tok[node] << 6) | c];
}

// ---------------- WMMA fp32 GEMM:  H[N,128] = X[N,K] @ W[K,128] ------------
// One wave computes one 16x16 tile of H.  Fragment layouts per CDNA5 ISA:
//   A (16x4 f32, 2 VGPR): lane = M (l&15), half-wave selects K pair (kb=2*(l>>4))
//   B (4x16 f32, 2 VGPR): lane = N (l&15), half-wave selects K pair
//   C/D (16x16 f32, 8 VGPR): vgpr j -> row j + 8*(l>>4), col = l&15
template<int K>
__global__ void gemm_wmma_kernel(const float* __restrict__ X,
                                 const float* __restrict__ W,
                                 float* __restrict__ H) {
  const int wave  = (blockIdx.x << 3) | (threadIdx.x >> 5);  // 8 waves/block
  const int lane  = threadIdx.x & 31;
  const int tileM = wave >> 3;       // 0 .. N_NODES/16-1
  const int tileN = wave & 7;        // 0 .. 7   (128 cols / 16)
  const int half  = lane >> 4;       // 0 | 1
  const int lid   = lane & 15;
  const int m     = tileM * 16 + lid;
  const int n     = tileN * 16 + lid;
  const int kb    = half * 2;

  v8f acc = {};
  const float* xrow = X + (size_t)m * K + kb;     // consecutive pair -> b64 load
  const float* wcol = W + (size_t)kb * 128 + n;   // column, stride 128
#pragma unroll
  for (int k = 0; k < K; k += 4) {
    v2f a; a.x = xrow[k];               a.y = xrow[k + 1];
    v2f b; b.x = wcol[(size_t)k * 128]; b.y = wcol[(size_t)(k + 1) * 128];
    acc = __builtin_amdgcn_wmma_f32_16x16x4_f32(
        /*neg_a=*/false, a, /*neg_b=*/false, b,
        /*c_mod=*/(short)0, acc, /*reuse_a=*/false, /*reuse_b=*/false);
  }

  float* hbase = H + ((size_t)(tileM * 16 + half * 8)) * 128 + tileN * 16 + lid;
#pragma unroll
  for (int j = 0; j < 8; ++j)
    hbase[(size_t)j * 128] = acc[j];
}

// ---------------- aggregation ----------------------------------------------

// out[i,c] = H[i,c] * dinv[i]^2   (self-loop term; also serves as zero-init)
__global__ void k_selfinit(const float* __restrict__ H, const float* __restrict__ dinv,
                           float* __restrict__ out) {
  size_t idx = (size_t)blockIdx.x * blockDim.x + threadIdx.x;   // < N*128
  int node = (int)(idx >> 7);
  float dv = dinv[node];
  out[idx] = H[idx] * dv * dv;
}

// out[dst] += H[src] * dinv[src]*dinv[dst]; 32 lanes per edge, float4 per lane
__global__ void k_scatter(const float* __restrict__ H, const int* __restrict__ src,
                          const int* __restrict__ dst, const float* __restrict__ dinv,
                          float* __restrict__ out) {
  size_t gid = (size_t)blockIdx.x * blockDim.x + threadIdx.x;
  int e    = (int)(gid >> 5);
  int lane = (int)(gid & 31);
  if (e >= N_EDGES) return;
  int s = src[e], d = dst[e];
  float nrm = dinv[s] * dinv[d];
  const float4 h = *(const float4*)(H + ((size_t)s << 7) + (lane << 2));
  float* o = out + ((size_t)d << 7) + (lane << 2);
  atomicAdd(o + 0, h.x * nrm);
  atomicAdd(o + 1, h.y * nrm);
  atomicAdd(o + 2, h.z * nrm);
  atomicAdd(o + 3, h.w * nrm);
}

// X[i,c] = relu(X[i,c] + b[c])   (in place)
__global__ void k_bias_relu(float* __restrict__ X, const float* __restrict__ b) {
  size_t idx = (size_t)blockIdx.x * blockDim.x + threadIdx.x;   // < N*128
  int c = (int)(idx & 127);
  float v = X[idx] + b[c];
  X[idx] = v > 0.0f ? v : 0.0f;
}

// ---------------- pooling + classifier -------------------------------------

__global__ void k_pool(const float* __restrict__ X, const int* __restrict__ batch,
                       float* __restrict__ sums, float* __restrict__ cnts) {
  size_t gid = (size_t)blockIdx.x * blockDim.x + threadIdx.x;
  int node = (int)(gid >> 5);
  int lane = (int)(gid & 31);
  if (node >= N_NODES) return;
  int g = batch[node];
  const float4 v = *(const float4*)(X + ((size_t)node << 7) + (lane << 2));
  float* o = sums + ((size_t)g << 7) + (lane << 2);
  atomicAdd(o + 0, v.x);
  atomicAdd(o + 1, v.y);
  atomicAdd(o + 2, v.z);
  atomicAdd(o + 3, v.w);
  if (lane == 0) atomicAdd(&cnts[g], 1.0f);
}

__global__ void k_final(const float* __restrict__ sums, const float* __restrict__ cnts,
                        const float* __restrict__ Wlin, const float* __restrict__ blin,
                        float* __restrict__ out) {
  int t = blockIdx.x * blockDim.x + threadIdx.x;
  if (t >= N_GRAPHS * NCLS) return;
  int g = t >> 2, c = t & 3;
  float inv = 1.0f / fmaxf(cnts[g], 1.0f);
  float acc = blin[c];
  const float* srow = sums + ((size_t)g << 7);
#pragma unroll 4
  for (int k = 0; k < 128; ++k)
    acc = fmaf(srow[k] * inv, Wlin[k * NCLS + c], acc);
  out[t] = acc;
}

// ---------------- driver ----------------------------------------------------

extern "C" void kernel_launch(void* const* d_in, const int* in_sizes, int n_in,
                              void* d_out, int out_size, void* d_ws, size_t ws_size,
                              hipStream_t stream) {
  const int*   tok   = (const int*)d_in[0];
  const int*   eidx  = (const int*)d_in[1];      // [2, E] row-major
  const int*   src   = eidx;
  const int*   dst   = eidx + N_EDGES;
  const int*   batch = (const int*)d_in[2];
  const float* emb   = (const float*)d_in[3];
  const float* W1    = (const float*)d_in[4];
  const float* b1    = (const float*)d_in[5];
  const float* W2    = (const float*)d_in[6];
  const float* b2    = (const float*)d_in[7];
  const float* Wlin  = (const float*)d_in[8];
  const float* blin  = (const float*)d_in[9];
  float*       out   = (float*)d_out;

  // workspace layout (floats): deg/dinv | x0[N,64] | A[N,128] | B[N,128] | sums | cnts
  float* ws   = (float*)d_ws;
  float* deg  = ws;
  float* x0   = deg  + N_NODES;
  float* A    = x0   + (size_t)N_NODES * 64;
  float* B    = A    + (size_t)N_NODES * 128;
  float* sums = B    + (size_t)N_NODES * 128;
  float* cnts = sums + (size_t)N_GRAPHS * 128;

  const int T = 256;
  const int gNodes  = (N_NODES + T - 1) / T;
  const int gEdges  = (N_EDGES + T - 1) / T;
  const int gEmbed  = (N_NODES * 64) / T;        // 25000, exact
  const int gFeat   = (N_NODES * 128) / T;       // 50000, exact
  const int gScat   = (N_EDGES * 32) / T;        // 75000, exact
  const int gPool   = (N_NODES * 32) / T;        // 12500, exact
  const int gGemm   = N_NODES / 16;              // 6250 blocks x 8 waves

  // normalization
  k_zero<<<gNodes, T, 0, stream>>>(deg, N_NODES);
  k_deg<<<gEdges, T, 0, stream>>>(dst, deg);
  k_dinv<<<gNodes, T, 0, stream>>>(deg);         // deg now holds dinv

  // embedding
  k_embed<<<gEmbed, T, 0, stream>>>(tok, emb, x0);

  // layer 1: h1 = x0 @ W1 ; x1 = relu(aggregate(h1) + b1)
  gemm_wmma_kernel<64><<<gGemm, T, 0, stream>>>(x0, W1, A);
  k_selfinit<<<gFeat, T, 0, stream>>>(A, deg, B);
  k_scatter<<<gScat, T, 0, stream>>>(A, src, dst, deg, B);
  k_bias_relu<<<gFeat, T, 0, stream>>>(B, b1);

  // layer 2: h2 = x1 @ W2 ; x2 = relu(aggregate(h2) + b2)
  gemm_wmma_kernel<128><<<gGemm, T, 0, stream>>>(B, W2, A);
  k_selfinit<<<gFeat, T, 0, stream>>>(A, deg, B);
  k_scatter<<<gScat, T, 0, stream>>>(A, src, dst, deg, B);
  k_bias_relu<<<gFeat, T, 0, stream>>>(B, b2);

  // mean pool + classifier
  k_zero<<<(N_GRAPHS * 128 + N_GRAPHS + T - 1) / T, T, 0, stream>>>(sums,
                                                    N_GRAPHS * 128 + N_GRAPHS);
  k_pool<<<gPool, T, 0, stream>>>(B, batch, sums, cnts);
  k_final<<<(N_GRAPHS * NCLS + T - 1) / T, T, 0, stream>>>(sums, cnts, Wlin, blin, out);
}